// VectorQuantizer_36799279792443
// MI455X (gfx1250) — compile-verified
//
#include <hip/hip_runtime.h>
#include <math.h>

typedef __attribute__((ext_vector_type(2))) float v2f;
typedef __attribute__((ext_vector_type(4))) float v4f;
typedef __attribute__((ext_vector_type(8))) float v8f;

constexpr int D = 256;              // embedding dim
constexpr int K = 1024;             // number of codes
constexpr int NROWS = 32 * 32 * 32; // 32768 flattened vectors
constexpr int ROWS_PER_WAVE = 16;
constexpr int WAVES_PER_BLOCK = 8;  // 256 threads / wave32
constexpr int LDS_STRIDE = 260;     // padded row stride (floats): conflict-free B reads

// ---------------------------------------------------------------------------
// Kernel 1: per-code squared norms, zero histogram + SSE accumulator
// ---------------------------------------------------------------------------
__global__ __launch_bounds__(256) void vq_init(const float* __restrict__ W,
                                               float* __restrict__ enorm,
                                               int* __restrict__ counts,
                                               double* __restrict__ sse) {
  int k = blockIdx.x * blockDim.x + threadIdx.x;
  if (k < K) {
    const v4f* w4 = (const v4f*)(W + (size_t)k * D);
    float s = 0.f;
#pragma unroll 4
    for (int i = 0; i < D / 4; ++i) {
      v4f v = w4[i];
      s += v.x * v.x + v.y * v.y + v.z * v.z + v.w * v.w;
    }
    enorm[k] = s;
    counts[k] = 0;
  }
  if (k == 0) *sse = 0.0;
}

// ---------------------------------------------------------------------------
// Kernel 2: fused distance GEMM (f32 WMMA 16x16x4) + argmin over K codes.
// One wave owns 16 rows. The whole block walks the 64 code tiles in lockstep:
// each tile's 16x256 weight slab is staged into a double-buffered LDS slot
// with CDNA5 async-to-LDS loads while the previous tile is being multiplied.
// ---------------------------------------------------------------------------
__global__ __launch_bounds__(32 * WAVES_PER_BLOCK) void vq_argmin(
    const float* __restrict__ X, const float* __restrict__ W,
    const float* __restrict__ enorm, int* __restrict__ idxArr,
    int* __restrict__ counts) {
  __shared__ float ldsW[2][16 * LDS_STRIDE];  // 2 x 16.25KB (+pad)

  const int tid = threadIdx.x;
  const int lane = tid & 31;
  const int wave = tid >> 5;
  const int half = lane >> 4;  // K-half selector for A/B fragments
  const int nrow = lane & 15;  // A row within tile == B column within tile
  const int baseRow = (blockIdx.x * WAVES_PER_BLOCK + wave) * ROWS_PER_WAVE;

  // A fragment source: X[baseRow+nrow][k0 + 2*half .. +1] (hot in WGP$)
  const float* xrow = X + (size_t)(baseRow + nrow) * D + 2 * half;

  // Async-stage one 16x256 weight tile into LDS buffer `buf`.
  // 1024 chunks of 16B; 256 threads -> 4 chunks each, ASYNCcnt-tracked.
  auto stage = [&](int nt, int buf) {
    const float* wbase = W + (size_t)nt * 16 * D;
    for (int j = tid; j < 1024; j += 256) {
      int r = j >> 6;           // weight row within tile
      int c = j & 63;           // 16-byte chunk within row
      unsigned ldsAddr =
          (unsigned)(uintptr_t)&ldsW[buf][r * LDS_STRIDE + c * 4];
      unsigned long long ga =
          (unsigned long long)(uintptr_t)(wbase + r * D + c * 4);
      asm volatile("global_load_async_to_lds_b128 %0, %1, off" ::"v"(ldsAddr),
                   "v"(ga)
                   : "memory");
    }
  };

  float minVal[8];
  int minIdx[8];
#pragma unroll
  for (int m = 0; m < 8; ++m) {
    minVal[m] = 3.4e38f;
    minIdx[m] = 0x7fffffff;
  }

  stage(0, 0);  // prologue: tile 0 in flight

  for (int nt = 0; nt < K / 16; ++nt) {
    const int buf = nt & 1;
    // My wave's async loads for tile nt are done; barrier makes it block-wide.
    asm volatile("s_wait_asynccnt 0" ::: "memory");
    __syncthreads();
    if (nt + 1 < K / 16) stage(nt + 1, buf ^ 1);  // overlap next tile

    const float* wlds = &ldsW[buf][nrow * LDS_STRIDE + 2 * half];

    v8f acc = {};
    v2f a0 = *(const v2f*)(xrow);
    v2f b0 = *(const v2f*)(wlds);
#pragma unroll
    for (int kk = 0; kk < D / 4; ++kk) {
      v2f a1 = a0, b1 = b0;
      if (kk + 1 < D / 4) {  // rotate: fetch next fragments before the MMA
        a1 = *(const v2f*)(xrow + (kk + 1) * 4);
        b1 = *(const v2f*)(wlds + (kk + 1) * 4);
      }
      acc = __builtin_amdgcn_wmma_f32_16x16x4_f32(
          /*neg_a=*/false, a0, /*neg_b=*/false, b0,
          /*c_mod=*/(short)0, acc, /*reuse_a=*/false, /*reuse_b=*/false);
      a0 = a1;
      b0 = b1;
    }

    const int code = nt * 16 + nrow;
    const float en = enorm[code];  // ||e||^2 for column N = nrow
#pragma unroll
    for (int m = 0; m < 8; ++m) {
      // element (M = m + 8*half, N = nrow): dist (minus row-constant ||x||^2)
      float dist = en - 2.0f * acc[m];
      if (dist < minVal[m]) {
        minVal[m] = dist;
        minIdx[m] = code;
      }
    }
    __syncthreads();  // everyone done reading buf before it is restaged
  }

  // Reduce across the 16 lanes of each half (XOR masks 1..8 stay in-half).
#pragma unroll
  for (int m = 0; m < 8; ++m) {
    float v = minVal[m];
    int id = minIdx[m];
#pragma unroll
    for (int off = 8; off >= 1; off >>= 1) {
      float ov = __shfl_xor(v, off, 32);
      int oid = __shfl_xor(id, off, 32);
      if (ov < v || (ov == v && oid < id)) {  // first-min tie-break
        v = ov;
        id = oid;
      }
    }
    if (nrow == 0) {  // lanes 0 and 16 publish rows m and m+8
      int row = baseRow + m + 8 * half;
      idxArr[row] = id;
      atomicAdd(&counts[id], 1);
    }
  }
}

// ---------------------------------------------------------------------------
// Kernel 3: per-row outputs: input copy, quantized gather, one-hot row,
// and SSE partial sums (quantized_ste == quantized numerically; the two
// losses are numerically identical means of (q - x)^2).
// ---------------------------------------------------------------------------
__global__ __launch_bounds__(256) void vq_scatter(
    const float* __restrict__ X, const float* __restrict__ W,
    const int* __restrict__ idxArr, float* __restrict__ outX,
    float* __restrict__ outQ, float* __restrict__ outE,
    double* __restrict__ sse) {
  __shared__ float red[256];
  const int row = blockIdx.x;
  const int t = threadIdx.x;
  const int id = idxArr[row];

  float x = X[(size_t)row * D + t];
  float w = W[(size_t)id * D + t];
  outX[(size_t)row * D + t] = x;
  outQ[(size_t)row * D + t] = w;
  float d = w - x;
  red[t] = d * d;

#pragma unroll
  for (int i = t; i < K; i += 256)
    outE[(size_t)row * K + i] = (i == id) ? 1.0f : 0.0f;

  __syncthreads();
  for (int s = 128; s > 0; s >>= 1) {
    if (t < s) red[t] += red[t + s];
    __syncthreads();
  }
  if (t == 0) atomicAdd(sse, (double)red[0]);
}

// ---------------------------------------------------------------------------
// Kernel 4: losses + perplexity
// ---------------------------------------------------------------------------
__global__ __launch_bounds__(1024) void vq_finalize(
    const int* __restrict__ counts, const double* __restrict__ sse,
    float* __restrict__ outScalars) {
  __shared__ float red[1024];
  int t = threadIdx.x;
  float p = (float)counts[t] / (float)NROWS;
  red[t] = p * logf(p + 1e-10f);
  __syncthreads();
  for (int s = 512; s > 0; s >>= 1) {
    if (t < s) red[t] += red[t + s];
    __syncthreads();
  }
  if (t == 0) {
    float mse = (float)(*sse / ((double)NROWS * (double)D));
    outScalars[0] = mse;            // q_latent_loss
    outScalars[1] = mse;            // e_latent_loss
    outScalars[2] = expf(-red[0]);  // perplexity
  }
}

// ---------------------------------------------------------------------------
extern "C" void kernel_launch(void* const* d_in, const int* in_sizes, int n_in,
                              void* d_out, int out_size, void* d_ws,
                              size_t ws_size, hipStream_t stream) {
  const float* X = (const float*)d_in[0];  // [32,32,32,256]
  const float* W = (const float*)d_in[1];  // [1024,256]
  float* out = (float*)d_out;

  char* ws = (char*)d_ws;
  float* enorm = (float*)ws;                           //   4 KB
  int* idxArr = (int*)(ws + 4096);                     // 128 KB
  int* counts = (int*)(ws + 4096 + 131072);            //   4 KB
  double* sse = (double*)(ws + 4096 + 131072 + 4096);  //   8 B

  float* outX = out;                         // inputs copy
  float* outQ = out + (size_t)NROWS * D;     // quantized_ste
  float* outE = out + (size_t)2 * NROWS * D; // one-hot encoding
  float* outS = out + (size_t)2 * NROWS * D + (size_t)NROWS * K;  // 3 scalars

  vq_init<<<(K + 255) / 256, 256, 0, stream>>>(W, enorm, counts, sse);
  vq_argmin<<<NROWS / (ROWS_PER_WAVE * WAVES_PER_BLOCK), 32 * WAVES_PER_BLOCK,
              0, stream>>>(X, W, enorm, idxArr, counts);
  vq_scatter<<<NROWS, 256, 0, stream>>>(X, W, idxArr, outX, outQ, outE, sse);
  vq_finalize<<<1, 1024, 0, stream>>>(counts, sse, outS);
}